// DeepSetsEnsemble_58145267253428
// MI455X (gfx1250) — compile-verified
//
#include <hip/hip_runtime.h>
#include <math.h>

// ---------------- problem constants ----------------
#define LPTS 524288
#define CIN  16
#define C1   128
#define C2   128
#define C3   64
#define NB   4096
#define DVEC 32
#define M1   256   // MLP hidden 1
#define M2   128   // MLP hidden 2
#define EPSV 1e-5f

// phi kernel tiling
#define WAVES 4            // waves per block
#define PHI_BLOCKS 1024    // persistent blocks (each loops over chunks)
#define S1 18              // LDS row stride for W1 (pad, even for b64 loads)
#define S2 130             // LDS row stride for W2
#define S3 130             // LDS row stride for W3

typedef float v2f __attribute__((ext_vector_type(2)));
typedef float v8f __attribute__((ext_vector_type(8)));

// D = A(16x4 f32) * B(4x16 f32) + C(16x16 f32)
__device__ __forceinline__ v8f wmma4(v2f a, v2f b, v8f c) {
  return __builtin_amdgcn_wmma_f32_16x16x4_f32(false, a, false, b, (short)0, c,
                                               false, false);
}

// ---------------- helper kernels ----------------
__global__ void zero_f32(float* __restrict__ p, int n) {
  int i = blockIdx.x * blockDim.x + threadIdx.x;
  if (i < n) p[i] = 0.0f;
}

// bounds[0]=0 ; bounds[j+1] = cumsum(counts)[j], counts = sample_indices[0,:]
__global__ __launch_bounds__(256) void scan_counts(const int* __restrict__ si,
                                                   int* __restrict__ bounds) {
  __shared__ int part[256];
  const int PER = NB / 256;  // 16
  int tid = threadIdx.x;
  int base = tid * PER;
  int loc[PER];
  int s = 0;
#pragma unroll
  for (int i = 0; i < PER; ++i) { s += si[base + i]; loc[i] = s; }
  part[tid] = s;
  __syncthreads();
  // Hillis-Steele inclusive scan of per-thread sums
  for (int off = 1; off < 256; off <<= 1) {
    int v = (tid >= off) ? part[tid - off] : 0;
    __syncthreads();
    part[tid] += v;
    __syncthreads();
  }
  int prefix = (tid == 0) ? 0 : part[tid - 1];
  if (tid == 0) bounds[0] = 0;
#pragma unroll
  for (int i = 0; i < PER; ++i) bounds[base + i + 1] = prefix + loc[i];
}

// ---------------- fused phi net + segment sums ----------------
// Persistent blocks: weights staged into LDS once per block, then each wave
// loops over 16-point columns running the 3-layer phi stack with
// V_WMMA_F32_16X16X4_F32, finishing with atomic segment accumulation.
__global__ __launch_bounds__(32 * WAVES) void phi_segsum(
    const float* __restrict__ xa,
    const float* __restrict__ W1, const float* __restrict__ b1,
    const float* __restrict__ W2, const float* __restrict__ b2,
    const float* __restrict__ W3, const float* __restrict__ b3,
    const int* __restrict__ bounds, float* __restrict__ sums) {
  extern __shared__ float smem[];
  float* sW1 = smem;                 // C1 x S1
  float* sW2 = sW1 + C1 * S1;        // C2 x S2
  float* sW3 = sW2 + C2 * S2;        // C3 x S3
  float* sB1 = sW3 + C3 * S3;        // C1
  float* sB2 = sB1 + C1;             // C2
  float* sB3 = sB2 + C2;             // C3
  float* sH  = sB3 + C3;             // WAVES x (C1*16) staging

  const int tid = threadIdx.x;
  const int nthr = 32 * WAVES;
  for (int i = tid; i < C1 * CIN; i += nthr) sW1[(i / CIN) * S1 + (i % CIN)] = W1[i];
  for (int i = tid; i < C2 * C1; i += nthr)  sW2[(i / C1) * S2 + (i % C1)] = W2[i];
  for (int i = tid; i < C3 * C2; i += nthr)  sW3[(i / C2) * S3 + (i % C2)] = W3[i];
  if (tid < C1) sB1[tid] = b1[tid];
  if (tid < C2) sB2[tid] = b2[tid];
  if (tid < C3) sB3[tid] = b3[tid];
  __syncthreads();

  const int wid  = tid >> 5;
  const int lane = tid & 31;
  const int col  = lane & 15;   // N / M-within-16 index
  const int hv   = lane >> 4;   // half-wave select
  float* hw = sH + wid * (C1 * 16);   // wave-private staging [128][16]
  const int total = bounds[NB];
  const int NCHUNK = LPTS / (16 * WAVES);

  for (int ch = blockIdx.x; ch < NCHUNK; ch += PHI_BLOCKS) {
    const int pbase = ch * (16 * WAVES) + wid * 16;

    // prefetch next chunk's x tile rows (speculative, hides the only cold loads)
    {
      int npb = pbase + PHI_BLOCKS * (16 * WAVES);
      if (npb < LPTS) {
#pragma unroll
        for (int r = 0; r < CIN; r += 4)
          __builtin_prefetch(&xa[(size_t)r * LPTS + npb + col], 0, 3);
      }
    }

    // ---- layer 1 B operand: x tile [16ch x 16pts] ----
    v2f bx[4];
#pragma unroll
    for (int k = 0; k < 4; ++k) {
      int r = 4 * k + 2 * hv;
      bx[k].x = xa[(size_t)r * LPTS + pbase + col];
      bx[k].y = xa[(size_t)(r + 1) * LPTS + pbase + col];
    }

    // ---- layer 1: h1[128x16] = relu(W1 @ x + b1) -> LDS ----
#pragma unroll
    for (int m = 0; m < C1 / 16; ++m) {
      v8f acc;
#pragma unroll
      for (int i = 0; i < 8; ++i) acc[i] = sB1[m * 16 + i + 8 * hv];
#pragma unroll
      for (int k = 0; k < 4; ++k) {
        v2f a = *(const v2f*)&sW1[(m * 16 + col) * S1 + 4 * k + 2 * hv];
        acc = wmma4(a, bx[k], acc);
      }
#pragma unroll
      for (int i = 0; i < 8; ++i) {
        float v = acc[i] > 0.f ? acc[i] : 0.f;
        hw[(m * 16 + i + 8 * hv) * 16 + col] = v;
      }
    }

    // ---- reload h1 as B operands (C-layout -> B-layout through LDS) ----
    v2f bt[32];
#pragma unroll
    for (int k = 0; k < 32; ++k) {
      int r = 4 * k + 2 * hv;
      bt[k].x = hw[r * 16 + col];
      bt[k].y = hw[(r + 1) * 16 + col];
    }

    // ---- layer 2: h2[128x16] = relu(W2 @ h1 + b2) -> LDS (overwrite) ----
#pragma unroll
    for (int m = 0; m < C2 / 16; ++m) {
      v8f acc;
#pragma unroll
      for (int i = 0; i < 8; ++i) acc[i] = sB2[m * 16 + i + 8 * hv];
#pragma unroll
      for (int k = 0; k < 32; ++k) {
        v2f a = *(const v2f*)&sW2[(m * 16 + col) * S2 + 4 * k + 2 * hv];
        acc = wmma4(a, bt[k], acc);
      }
#pragma unroll
      for (int i = 0; i < 8; ++i) {
        float v = acc[i] > 0.f ? acc[i] : 0.f;
        hw[(m * 16 + i + 8 * hv) * 16 + col] = v;
      }
    }

    // ---- reload h2 as B operands ----
#pragma unroll
    for (int k = 0; k < 32; ++k) {
      int r = 4 * k + 2 * hv;
      bt[k].x = hw[r * 16 + col];
      bt[k].y = hw[(r + 1) * 16 + col];
    }

    // ---- layer 3: h3[64x16] = relu(W3 @ h2 + b3) kept in registers ----
    v8f h3[C3 / 16];
#pragma unroll
    for (int m = 0; m < C3 / 16; ++m) {
      v8f acc;
#pragma unroll
      for (int i = 0; i < 8; ++i) acc[i] = sB3[m * 16 + i + 8 * hv];
#pragma unroll
      for (int k = 0; k < 32; ++k) {
        v2f a = *(const v2f*)&sW3[(m * 16 + col) * S3 + 4 * k + 2 * hv];
        acc = wmma4(a, bt[k], acc);
      }
#pragma unroll
      for (int i = 0; i < 8; ++i) acc[i] = acc[i] > 0.f ? acc[i] : 0.f;
      h3[m] = acc;
    }

    // ---- segment accumulation: lane owns point p ----
    int p = pbase + col;
    int lo = 0, hi = NB;  // invariant: bounds[lo] <= p < bounds[hi]
#pragma unroll
    for (int it = 0; it < 12; ++it) {  // log2(4096)
      int mid = (lo + hi) >> 1;
      if (bounds[mid] <= p) lo = mid; else hi = mid;
    }
    if (p < total) {
#pragma unroll
      for (int m = 0; m < C3 / 16; ++m)
#pragma unroll
        for (int i = 0; i < 8; ++i)
          atomicAdd(&sums[(size_t)lo * C3 + m * 16 + i + 8 * hv], h3[m][i]);
    }
  }
}

// ---------------- z = [means | x_b] ----------------
__global__ void build_z(const float* __restrict__ sums, const int* __restrict__ si,
                        const float* __restrict__ xb, float* __restrict__ z) {
  int idx = blockIdx.x * blockDim.x + threadIdx.x;
  if (idx >= NB * (C3 + DVEC)) return;
  int b = idx / (C3 + DVEC), f = idx % (C3 + DVEC);
  float v;
  if (f < C3) {
    int c = si[b];  // counts row 0
    v = sums[(size_t)b * C3 + f] / (c > 0 ? (float)c : 1.0f);
  } else {
    v = xb[(size_t)b * DVEC + (f - C3)];
  }
  z[idx] = v;
}

// ---------------- WMMA GEMM: Y[M,N] = A[M,K] @ W[N,K]^T + bias ----------------
// also atomically accumulates per-column sum / sum-of-squares for BatchNorm.
__global__ __launch_bounds__(256) void gemm_bias_stats(
    const float* __restrict__ A, const float* __restrict__ W,
    const float* __restrict__ bias, float* __restrict__ Y,
    float* __restrict__ cs, float* __restrict__ cq, int M, int N, int K) {
  int gw = (blockIdx.x * blockDim.x + threadIdx.x) >> 5;  // global wave id
  int lane = threadIdx.x & 31;
  int col = lane & 15, hv = lane >> 4;
  int ntn = N >> 4;
  int tn = gw % ntn, tm = gw / ntn;
  if (tm >= (M >> 4)) return;  // wave-uniform
  int r0 = tm * 16, c0 = tn * 16;

  v8f acc;
  float bn = bias[c0 + col];
#pragma unroll
  for (int i = 0; i < 8; ++i) acc[i] = bn;

  const float* arow = A + (size_t)(r0 + col) * K + 2 * hv;
  const float* brow = W + (size_t)(c0 + col) * K + 2 * hv;
  for (int k = 0; k < K; k += 4) {
    v2f a = *(const v2f*)(arow + k);
    v2f b = *(const v2f*)(brow + k);
    acc = wmma4(a, b, acc);
  }

  float s = 0.f, s2 = 0.f;
#pragma unroll
  for (int i = 0; i < 8; ++i) {
    float v = acc[i];
    Y[(size_t)(r0 + i + 8 * hv) * N + c0 + col] = v;
    s += v; s2 += v * v;
  }
  atomicAdd(&cs[c0 + col], s);
  atomicAdd(&cq[c0 + col], s2);
}

// ---------------- BN (training stats, biased var) + ReLU, in place ----------------
__global__ void bn_relu(float* __restrict__ Y, const float* __restrict__ cs,
                        const float* __restrict__ cq, const float* __restrict__ g,
                        const float* __restrict__ be, int M, int N) {
  int idx = blockIdx.x * blockDim.x + threadIdx.x;
  if (idx >= M * N) return;
  int c = idx % N;
  float invm = 1.0f / (float)M;
  float mu = cs[c] * invm;
  float var = cq[c] * invm - mu * mu;
  float v = (Y[idx] - mu) * rsqrtf(var + EPSV) * g[c] + be[c];
  Y[idx] = v > 0.f ? v : 0.f;
}

// ---------------- final dot + sigmoid ----------------
__global__ void final_sig(const float* __restrict__ Y, const float* __restrict__ W,
                          const float* __restrict__ b, float* __restrict__ out,
                          int M, int K) {
  int r = blockIdx.x * blockDim.x + threadIdx.x;
  if (r >= M) return;
  float s = b[0];
  const float* row = Y + (size_t)r * K;
  for (int k = 0; k < K; ++k) s += row[k] * W[k];
  out[r] = 1.0f / (1.0f + __expf(-s));
}

// ---------------- launch ----------------
extern "C" void kernel_launch(void* const* d_in, const int* in_sizes, int n_in,
                              void* d_out, int out_size, void* d_ws, size_t ws_size,
                              hipStream_t stream) {
  const float* xa  = (const float*)d_in[0];
  const float* xb  = (const float*)d_in[1];
  const int*   si  = (const int*)d_in[2];
  const float* W1  = (const float*)d_in[3];  const float* b1  = (const float*)d_in[4];
  const float* W2  = (const float*)d_in[5];  const float* b2  = (const float*)d_in[6];
  const float* W3  = (const float*)d_in[7];  const float* b3  = (const float*)d_in[8];
  const float* Wm1 = (const float*)d_in[9];  const float* bm1 = (const float*)d_in[10];
  const float* g1  = (const float*)d_in[11]; const float* be1 = (const float*)d_in[12];
  const float* Wm2 = (const float*)d_in[13]; const float* bm2 = (const float*)d_in[14];
  const float* g2  = (const float*)d_in[15]; const float* be2 = (const float*)d_in[16];
  const float* Wm3 = (const float*)d_in[17]; const float* bm3 = (const float*)d_in[18];

  float* ws   = (float*)d_ws;
  float* sums = ws;                             // NB*C3
  float* z    = sums + (size_t)NB * C3;         // NB*96
  float* y1   = z + (size_t)NB * (C3 + DVEC);   // NB*M1
  float* y2   = y1 + (size_t)NB * M1;           // NB*M2
  float* cs1  = y2 + (size_t)NB * M2;           // M1
  float* cq1  = cs1 + M1;                       // M1
  float* cs2  = cq1 + M1;                       // M2
  float* cq2  = cs2 + M2;                       // M2
  int*   bounds = (int*)(cq2 + M2);             // NB+1 ints

  // init accumulators (harness does not re-zero between replays)
  zero_f32<<<(NB * C3 + 255) / 256, 256, 0, stream>>>(sums, NB * C3);
  zero_f32<<<(2 * M1 + 2 * M2 + 255) / 256, 256, 0, stream>>>(cs1, 2 * M1 + 2 * M2);
  scan_counts<<<1, 256, 0, stream>>>(si, bounds);

  size_t shmem = (size_t)(C1 * S1 + C2 * S2 + C3 * S3 + C1 + C2 + C3 +
                          WAVES * C1 * 16) * sizeof(float);
  hipFuncSetAttribute(reinterpret_cast<const void*>(phi_segsum),
                      hipFuncAttributeMaxDynamicSharedMemorySize, (int)shmem);
  phi_segsum<<<PHI_BLOCKS, 32 * WAVES, shmem, stream>>>(
      xa, W1, b1, W2, b2, W3, b3, bounds, sums);

  build_z<<<(NB * (C3 + DVEC) + 255) / 256, 256, 0, stream>>>(sums, si, xb, z);

  gemm_bias_stats<<<(NB / 16) * (M1 / 16) / 8, 256, 0, stream>>>(
      z, Wm1, bm1, y1, cs1, cq1, NB, M1, C3 + DVEC);
  bn_relu<<<(NB * M1 + 255) / 256, 256, 0, stream>>>(y1, cs1, cq1, g1, be1, NB, M1);

  gemm_bias_stats<<<(NB / 16) * (M2 / 16) / 8, 256, 0, stream>>>(
      y1, Wm2, bm2, y2, cs2, cq2, NB, M2, M1);
  bn_relu<<<(NB * M2 + 255) / 256, 256, 0, stream>>>(y2, cs2, cq2, g2, be2, NB, M2);

  final_sig<<<(NB + 255) / 256, 256, 0, stream>>>(y2, Wm3, bm3, (float*)d_out, NB, M2);
}